// GCN_71313636982801
// MI455X (gfx1250) — compile-verified
//
#include <hip/hip_runtime.h>
#include <hip/hip_bf16.h>
#include <math.h>

// ---------------------------------------------------------------------------
// 2-layer GCN on gfx1250:
//   deg/norm -> WMMA GEMM1 (+ fused self-loop seed) -> edge scatter (f32
//   atomics, L2-resident) -> WMMA GEMM2 (bias+ReLU fused into A-load, fused
//   self-loop seed into d_out) -> edge scatter -> log_softmax.
// fp32 WMMA (V_WMMA_F32_16X16X4_F32) keeps full precision for validation.
// ---------------------------------------------------------------------------

typedef float v2f __attribute__((ext_vector_type(2)));
typedef float v8f __attribute__((ext_vector_type(8)));

#define F_IN 128
#define HID  16
#define NCLS 8

// ---- degree / norm ---------------------------------------------------------

__global__ void deg_init_kernel(float* __restrict__ deg, int n) {
  int i = blockIdx.x * blockDim.x + threadIdx.x;
  if (i < n) deg[i] = 1.0f;  // self-loop contributes 1 to every node
}

__global__ void deg_count_kernel(const int* __restrict__ dst,
                                 float* __restrict__ deg, int e) {
  int i = blockIdx.x * blockDim.x + threadIdx.x;
  if (i < e) atomicAdd(&deg[dst[i]], 1.0f);
}

__global__ void dinv_kernel(float* __restrict__ deg, int n) {
  int i = blockIdx.x * blockDim.x + threadIdx.x;
  if (i < n) deg[i] = rsqrtf(deg[i]);  // deg >= 1 always (self-loops)
}

// ---- GEMM1: h = x @ W1  (N x 128)@(128 x 16), fp32 WMMA --------------------
// One wave owns one 16-row tile; 32 WMMA steps of K=4.
// A 16x4: lane<16 holds M=lane {K0,K1}; lane>=16 holds M=lane-16 {K2,K3}.
// B 4x16: lane<16 holds N=lane {K0,K1}; lane>=16 holds N=lane-16 {K2,K3}.
// C/D:    vgpr r, lane<16 -> (M=r, N=lane); lane>=16 -> (M=r+8, N=lane-16).
// Store fuses the self-loop seed: out1 = h * dinv^2 (dinv spread via __shfl).

__global__ void __launch_bounds__(256) gemm1_kernel(
    const float* __restrict__ x, const float* __restrict__ W1,
    const float* __restrict__ dinv,
    float* __restrict__ h, float* __restrict__ out1, int nTiles, int n) {
  int wave = (blockIdx.x * 256 + threadIdx.x) >> 5;  // uniform within wave
  int lane = threadIdx.x & 31;
  if (wave >= nTiles) return;  // wave-uniform exit: EXEC all-1s for WMMA
  int half = lane >> 4;
  int m    = lane & 15;
  int row  = wave * 16 + m;
  int rowc = row < n ? row : n - 1;  // clamped load (no divergence)
  const float* xrow = x + (size_t)rowc * F_IN;

  v8f c = {};
  #pragma unroll 4
  for (int k0 = 0; k0 < F_IN; k0 += 4) {
    int kk = k0 + 2 * half;
    v2f a = *(const v2f*)(xrow + kk);   // A[row][kk], A[row][kk+1]
    v2f b;
    b.x = W1[kk * HID + m];             // B[kk][m]
    b.y = W1[(kk + 1) * HID + m];       // B[kk+1][m]
    c = __builtin_amdgcn_wmma_f32_16x16x4_f32(
        false, a, false, b, (short)0, c, false, false);
  }

  // lane j (j=lane&15) holds dinv for tile-row j; spread across C rows by shfl
  float dv = dinv[rowc];
  bool full = (wave * 16 + 16 <= n);  // wave-uniform
  if (full) {
    #pragma unroll
    for (int r = 0; r < 8; ++r) {
      int orow = wave * 16 + r + 8 * half;
      float w  = __shfl(dv, r + 8 * half, 32);
      size_t o = (size_t)orow * HID + m;
      h[o]    = c[r];
      out1[o] = c[r] * w * w;
    }
  } else {
    #pragma unroll
    for (int r = 0; r < 8; ++r) {
      int orow = wave * 16 + r + 8 * half;
      float w  = __shfl(dv, r + 8 * half, 32);
      if (orow < n) {
        size_t o = (size_t)orow * HID + m;
        h[o]    = c[r];
        out1[o] = c[r] * w * w;
      }
    }
  }
}

// ---- edge scatter, layer 1: out1[d] += h[s] * dinv[s]*dinv[d] --------------

__global__ void scatter1_kernel(const int* __restrict__ src,
                                const int* __restrict__ dst,
                                const float* __restrict__ dinv,
                                const float* __restrict__ h,
                                float* __restrict__ out1, int e) {
  int i = blockIdx.x * blockDim.x + threadIdx.x;
  if (i >= e) return;
  int s = src[i];
  int d = dst[i];
  float w = dinv[s] * dinv[d];
  const float4* hp = (const float4*)(h + (size_t)s * HID);  // 64B-aligned row
  float* op = out1 + (size_t)d * HID;
  #pragma unroll
  for (int q = 0; q < 4; ++q) {
    float4 v = hp[q];
    atomicAdd(op + q * 4 + 0, v.x * w);
    atomicAdd(op + q * 4 + 1, v.y * w);
    atomicAdd(op + q * 4 + 2, v.z * w);
    atomicAdd(op + q * 4 + 3, v.w * w);
  }
}

// ---- GEMM2: lt = relu(out1 + b1) @ W2  (N x 16)@(16 x 8), fp32 WMMA --------
// bias+ReLU fused into the A-operand load (K dim == hidden channel dim).
// B zero-padded to 16 cols via value masking (loads stay uniform, EXEC all-1s).
// Store fuses the layer-2 self-loop seed into d_out: out = lt * dinv^2.

__global__ void __launch_bounds__(256) gemm2_kernel(
    const float* __restrict__ h1, const float* __restrict__ b1,
    const float* __restrict__ W2, const float* __restrict__ dinv,
    float* __restrict__ lt, float* __restrict__ out, int nTiles, int n) {
  int wave = (blockIdx.x * 256 + threadIdx.x) >> 5;
  int lane = threadIdx.x & 31;
  if (wave >= nTiles) return;
  int half = lane >> 4;
  int m    = lane & 15;
  int row  = wave * 16 + m;
  int rowc = row < n ? row : n - 1;
  const float* hrow = h1 + (size_t)rowc * HID;
  float maskv = (m < NCLS) ? 1.0f : 0.0f;
  int mc = m & (NCLS - 1);

  v8f c = {};
  #pragma unroll
  for (int k0 = 0; k0 < HID; k0 += 4) {
    int kk = k0 + 2 * half;
    v2f a;
    a.x = fmaxf(hrow[kk]     + b1[kk],     0.0f);  // fused bias + ReLU
    a.y = fmaxf(hrow[kk + 1] + b1[kk + 1], 0.0f);
    v2f b;
    b.x = W2[kk * NCLS + mc] * maskv;
    b.y = W2[(kk + 1) * NCLS + mc] * maskv;
    c = __builtin_amdgcn_wmma_f32_16x16x4_f32(
        false, a, false, b, (short)0, c, false, false);
  }

  float dv = dinv[rowc];
  bool full = (wave * 16 + 16 <= n);
  if (m < NCLS) {
    #pragma unroll
    for (int r = 0; r < 8; ++r) {
      int orow = wave * 16 + r + 8 * half;
      float w  = __shfl(dv, r + 8 * half, 32);
      if (full || orow < n) {
        size_t o = (size_t)orow * NCLS + m;
        lt[o]  = c[r];
        out[o] = c[r] * w * w;   // self-loop seed of layer-2 aggregation
      }
    }
  }
}

// ---- edge scatter, layer 2: out[d] += lt[s] * dinv[s]*dinv[d] --------------

__global__ void scatter2_kernel(const int* __restrict__ src,
                                const int* __restrict__ dst,
                                const float* __restrict__ dinv,
                                const float* __restrict__ lt,
                                float* __restrict__ out, int e) {
  int i = blockIdx.x * blockDim.x + threadIdx.x;
  if (i >= e) return;
  int s = src[i];
  int d = dst[i];
  float w = dinv[s] * dinv[d];
  const float4* lp = (const float4*)(lt + (size_t)s * NCLS);  // 32B-aligned
  float* op = out + (size_t)d * NCLS;
  #pragma unroll
  for (int q = 0; q < 2; ++q) {
    float4 v = lp[q];
    atomicAdd(op + q * 4 + 0, v.x * w);
    atomicAdd(op + q * 4 + 1, v.y * w);
    atomicAdd(op + q * 4 + 2, v.z * w);
    atomicAdd(op + q * 4 + 3, v.w * w);
  }
}

// ---- final: add b2, log_softmax over 8 classes, in place on d_out ----------

__global__ void lsm_kernel(float* __restrict__ out,
                           const float* __restrict__ b2, int n) {
  int i = blockIdx.x * blockDim.x + threadIdx.x;
  if (i >= n) return;
  float* row = out + (size_t)i * NCLS;
  float z[NCLS];
  float mx = -INFINITY;
  #pragma unroll
  for (int c = 0; c < NCLS; ++c) {
    z[c] = row[c] + b2[c];
    mx = fmaxf(mx, z[c]);
  }
  float s = 0.0f;
  #pragma unroll
  for (int c = 0; c < NCLS; ++c) s += expf(z[c] - mx);
  float l = mx + logf(s);
  #pragma unroll
  for (int c = 0; c < NCLS; ++c) row[c] = z[c] - l;
}

// ---------------------------------------------------------------------------

extern "C" void kernel_launch(void* const* d_in, const int* in_sizes, int n_in,
                              void* d_out, int out_size, void* d_ws, size_t ws_size,
                              hipStream_t stream) {
  const float* x  = (const float*)d_in[0];
  const int* edge = (const int*)d_in[1];   // int32 on device (JAX x64 off)
  const float* W1 = (const float*)d_in[2];
  const float* b1 = (const float*)d_in[3];
  const float* W2 = (const float*)d_in[4];
  const float* b2 = (const float*)d_in[5];

  int n = in_sizes[0] / F_IN;   // 100000
  int e = in_sizes[1] / 2;      // 3200000
  const int* src = edge;        // edge_index[0]
  const int* dst = edge + e;    // edge_index[1]

  // workspace: deg/dinv (n) | h (n*16) | h1/out1 (n*16) | lt (n*8)  ~16.4 MB
  float* deg = (float*)d_ws;
  float* h   = deg + n;
  float* h1  = h + (size_t)n * HID;
  float* lt  = h1 + (size_t)n * HID;
  float* out = (float*)d_out;

  const int T = 256;
  int nTiles  = (n + 15) / 16;
  int gBlocks = (nTiles + 7) / 8;  // 8 waves per 256-thread block

  deg_init_kernel<<<(n + T - 1) / T, T, 0, stream>>>(deg, n);
  deg_count_kernel<<<(e + T - 1) / T, T, 0, stream>>>(dst, deg, e);
  dinv_kernel<<<(n + T - 1) / T, T, 0, stream>>>(deg, n);

  // GEMM1 writes h and the self-loop-seeded accumulator h1
  gemm1_kernel<<<gBlocks, T, 0, stream>>>(x, W1, deg, h, h1, nTiles, n);
  scatter1_kernel<<<(e + T - 1) / T, T, 0, stream>>>(src, dst, deg, h, h1, e);

  // GEMM2 applies bias+ReLU on the fly, writes lt and the seeded d_out
  gemm2_kernel<<<gBlocks, T, 0, stream>>>(h1, b1, W2, deg, lt, out, nTiles, n);
  scatter2_kernel<<<(e + T - 1) / T, T, 0, stream>>>(src, dst, deg, lt, out, e);

  lsm_kernel<<<(n + T - 1) / T, T, 0, stream>>>(out, b2, n);
}